// ABCNN_2_18056042512596
// MI455X (gfx1250) — compile-verified
//
#include <hip/hip_runtime.h>
#include <hip/hip_bf16.h>
#include <math.h>

// ---------------- problem constants ----------------
#define BB   64          // batch
#define LL   512         // sequence length
#define DD   512         // feature dim
#define WW   4           // conv window
#define LC   515         // conv output length = LL + WW - 1
#define TILE 128         // output tile (i and j) per workgroup
#define KT   32          // K slab staged in LDS
#define LSTR 36          // LDS row stride in floats (16B-aligned rows, conflict-free)

typedef __attribute__((ext_vector_type(2))) float v2f;
typedef __attribute__((ext_vector_type(8))) float v8f;

// GCC-vector int4: matches the async-LDS builtin's parameter pointee type
typedef int b128_t __attribute__((vector_size(16)));
typedef __attribute__((address_space(1))) b128_t* gas_b128_t;
typedef __attribute__((address_space(3))) b128_t* las_b128_t;

// ---------------- async global->LDS copy helpers (gfx1250) ----------------
__device__ __forceinline__ void async_ld_b128(const float* g, float* l) {
#if __has_builtin(__builtin_amdgcn_global_load_async_to_lds_b128)
    __builtin_amdgcn_global_load_async_to_lds_b128(
        (gas_b128_t)g, (las_b128_t)l, 0, 0);
#else
    unsigned lofs = (unsigned)(__UINTPTR_TYPE__)(__attribute__((address_space(3))) float*)l;
    asm volatile("global_load_async_to_lds_b128 %0, %1, off"
                 :: "v"(lofs), "v"(g)
                 : "memory");
#endif
}

__device__ __forceinline__ void wait_async_le8() {
#if __has_builtin(__builtin_amdgcn_s_wait_asynccnt)
    __builtin_amdgcn_s_wait_asynccnt(8);
#else
    asm volatile("s_wait_asynccnt 0x8" ::: "memory");
#endif
}

__device__ __forceinline__ void wait_async_le0() {
#if __has_builtin(__builtin_amdgcn_s_wait_asynccnt)
    __builtin_amdgcn_s_wait_asynccnt(0);
#else
    asm volatile("s_wait_asynccnt 0x0" ::: "memory");
#endif
}

// ---------------- zero the gate accumulators ----------------
__global__ void zero_kernel(float* p, int n) {
    int i = blockIdx.x * blockDim.x + threadIdx.x;
    if (i < n) p[i] = 0.0f;
}

// ---------------- fused mask * conv * tanh + row squared norms ----------------
// grid: (LC, B, 2), block: 256
__global__ void conv_tanh_kernel(const float* __restrict__ F0,
                                 const float* __restrict__ F1,
                                 const float* __restrict__ m0,
                                 const float* __restrict__ m1,
                                 const float* __restrict__ cw,
                                 const float* __restrict__ cb,
                                 float* __restrict__ c0, float* __restrict__ c1,
                                 float* __restrict__ sq0, float* __restrict__ sq1) {
    const int i     = blockIdx.x;   // conv output row, 0..LC-1
    const int b     = blockIdx.y;   // batch
    const int which = blockIdx.z;   // 0 -> tensor0, 1 -> tensor1

    const float* __restrict__ F = which ? F1 : F0;
    const float* __restrict__ M = which ? m1 : m0;
    float* __restrict__ C  = which ? c1  : c0;
    float* __restrict__ SQ = which ? sq1 : sq0;

    const float w0 = cw[0], w1 = cw[1], w2 = cw[2], w3 = cw[3];
    const float bias = cb[0];

    __shared__ float red[256];
    float sqacc = 0.0f;

    // cross-correlation: y[i] = bias + sum_k w[k] * x[i - (W-1) + k]
    for (int d = threadIdx.x; d < DD; d += blockDim.x) {
        float acc = bias;
#pragma unroll
        for (int k = 0; k < WW; ++k) {
            int l = i - (WW - 1) + k;
            if (l >= 0 && l < LL) {
                float wk = (k == 0) ? w0 : (k == 1) ? w1 : (k == 2) ? w2 : w3;
                acc += wk * F[((size_t)b * LL + l) * DD + d] * M[b * LL + l];
            }
        }
        float c = tanhf(acc);
        C[((size_t)b * LC + i) * DD + d] = c;
        sqacc += c * c;
    }

    red[threadIdx.x] = sqacc;
    __syncthreads();
#pragma unroll
    for (int s = 128; s > 0; s >>= 1) {
        if (threadIdx.x < (unsigned)s) red[threadIdx.x] += red[threadIdx.x + s];
        __syncthreads();
    }
    if (threadIdx.x == 0) SQ[b * LC + i] = red[0];
}

// ---------------- batched fp32 WMMA match-score + row/col sums ----------------
// grid: (ceil(LC/TILE), ceil(LC/TILE), B), block: 256 (8 waves)

// Issue one K-slab fill: 8 async B128 loads per thread (4 for A, 4 for B).
// Out-of-range rows are clamped (not zeroed): they only feed (i,j) elements
// that the epilogue masks to zero, and all data is finite (tanh output).
__device__ __forceinline__ void async_fill(const float* __restrict__ Cb0,
                                           const float* __restrict__ Cb1,
                                           float* __restrict__ ldsA,
                                           float* __restrict__ ldsB,
                                           int tid, int i0, int j0, int k0) {
#pragma unroll
    for (int r = 0; r < 4; ++r) {
        int f   = tid + 256 * r;     // 0..1023 float4 id
        int row = f >> 3;            // 8 float4 per 32-float row
        int cf  = (f & 7) * 4;
        int gr  = i0 + row; if (gr > LC - 1) gr = LC - 1;
        int gc  = j0 + row; if (gc > LC - 1) gc = LC - 1;
        async_ld_b128(Cb0 + (size_t)gr * DD + k0 + cf, ldsA + row * LSTR + cf);
        async_ld_b128(Cb1 + (size_t)gc * DD + k0 + cf, ldsB + row * LSTR + cf);
    }
}

__global__ __launch_bounds__(256)
void match_kernel(const float* __restrict__ c0, const float* __restrict__ c1,
                  const float* __restrict__ sq0, const float* __restrict__ sq1,
                  float* __restrict__ a0, float* __restrict__ a1) {
    __shared__ float As[2][TILE * LSTR];
    __shared__ float Bs[2][TILE * LSTR];
    __shared__ float sq0s[TILE], sq1s[TILE];
    __shared__ float a0s[TILE], a1s[TILE];

    const int b  = blockIdx.z;
    const int i0 = blockIdx.x * TILE;
    const int j0 = blockIdx.y * TILE;
    const int tid  = threadIdx.x;
    const int lane = tid & 31;
    const int wave = tid >> 5;       // 0..7
    const int wrow = wave & 3;       // 4 wave-rows * 32 rows each
    const int wcol = wave >> 2;      // 2 wave-cols * 64 cols each
    const int nloc = lane & 15;
    const int half = lane >> 4;      // 0/1 -> m offset 0/8 in C layout

    const float* __restrict__ Cb0 = c0 + (size_t)b * LC * DD;
    const float* __restrict__ Cb1 = c1 + (size_t)b * LC * DD;

    // kick off the first K slab while we set up
    async_fill(Cb0, Cb1, As[0], Bs[0], tid, i0, j0, 0);

    if (tid < TILE) {
        int gi = i0 + tid, gj = j0 + tid;
        sq0s[tid] = (gi < LC) ? sq0[b * LC + gi] : 0.0f;
        sq1s[tid] = (gj < LC) ? sq1[b * LC + gj] : 0.0f;
        a0s[tid] = 0.0f;
        a1s[tid] = 0.0f;
    }

    v8f acc[2][4];
#pragma unroll
    for (int ms = 0; ms < 2; ++ms)
#pragma unroll
        for (int ns = 0; ns < 4; ++ns)
            acc[ms][ns] = {};

    const int NS = DD / KT;          // 16 K slabs
    for (int s = 0; s < NS; ++s) {
        const float* curA = As[s & 1];
        const float* curB = Bs[s & 1];
        if (s + 1 < NS) {
            // prefetch next slab into the other buffer (its last readers
            // were barriered at the end of iteration s-1)
            async_fill(Cb0, Cb1, As[(s + 1) & 1], Bs[(s + 1) & 1],
                       tid, i0, j0, (s + 1) * KT);
            wait_async_le8();        // this wave's slab-s loads retired
        } else {
            wait_async_le0();
        }
        __syncthreads();             // all waves' slab-s data visible in LDS

#pragma unroll
        for (int kk = 0; kk < KT; kk += 4) {
            // A 16x4 layout: lanes 0-15 hold K=k..k+1, lanes 16-31 hold K=k+2..k+3
            int klane = kk + 2 * half;
            v2f afrag[2], bfrag[4];
#pragma unroll
            for (int ms = 0; ms < 2; ++ms) {
                int row = wrow * 32 + ms * 16 + nloc;
                afrag[ms] = *(const v2f*)(&curA[row * LSTR + klane]);
            }
#pragma unroll
            for (int ns = 0; ns < 4; ++ns) {
                int col = wcol * 64 + ns * 16 + nloc;
                bfrag[ns] = *(const v2f*)(&curB[col * LSTR + klane]);
            }
#pragma unroll
            for (int ms = 0; ms < 2; ++ms)
#pragma unroll
                for (int ns = 0; ns < 4; ++ns)
                    acc[ms][ns] = __builtin_amdgcn_wmma_f32_16x16x4_f32(
                        false, afrag[ms], false, bfrag[ns],
                        (short)0, acc[ms][ns], false, false);
        }
        __syncthreads();             // slab-s buffer free for prefetch s+2
    }

    // Epilogue: score = 1/(1+sqrt(max(sq0_i + sq1_j - 2*cross, 1e-12))); gate sums.
    float a1part[4] = {0.f, 0.f, 0.f, 0.f};
#pragma unroll
    for (int ms = 0; ms < 2; ++ms) {
#pragma unroll
        for (int v = 0; v < 8; ++v) {
            int lm = wrow * 32 + ms * 16 + v + 8 * half;  // local i
            int gi = i0 + lm;
            float rowpart = 0.0f;
#pragma unroll
            for (int ns = 0; ns < 4; ++ns) {
                int ln = wcol * 64 + ns * 16 + nloc;      // local j
                int gj = j0 + ln;
                float cross = acc[ms][ns][v];
                float d2 = sq0s[lm] + sq1s[ln] - 2.0f * cross;
                float dist = sqrtf(fmaxf(d2, 1e-12f));
                float sc = 1.0f / (1.0f + dist);
                if (gi >= LC || gj >= LC) sc = 0.0f;
                rowpart += sc;
                a1part[ns] += sc;
            }
            atomicAdd(&a0s[lm], rowpart);
        }
    }
#pragma unroll
    for (int ns = 0; ns < 4; ++ns) {
        int ln = wcol * 64 + ns * 16 + nloc;
        atomicAdd(&a1s[ln], a1part[ns]);
    }
    __syncthreads();

    if (tid < TILE) {
        int gi = i0 + tid;
        if (gi < LC) atomicAdd(&a0[b * LC + gi], a0s[tid]);
        int gj = j0 + tid;
        if (gj < LC) atomicAdd(&a1[b * LC + gj], a1s[tid]);
    }
}

// ---------------- gated average-pool output ----------------
// grid: (LL, B, 2), block: 256
__global__ void pool_out_kernel(const float* __restrict__ c0,
                                const float* __restrict__ c1,
                                const float* __restrict__ a0,
                                const float* __restrict__ a1,
                                float* __restrict__ out) {
    const int t     = blockIdx.x;    // 0..LL-1 (pool output index)
    const int b     = blockIdx.y;
    const int which = blockIdx.z;

    const float* __restrict__ C = which ? c1 : c0;
    const float* __restrict__ A = which ? a1 : a0;
    float* __restrict__ O = out + (size_t)which * BB * LL * DD;

    for (int d = threadIdx.x; d < DD; d += blockDim.x) {
        float s = 0.0f;
#pragma unroll
        for (int u = 0; u < WW; ++u) {
            s += C[((size_t)b * LC + t + u) * DD + d] * A[b * LC + t + u];
        }
        O[((size_t)b * LL + t) * DD + d] = s * (1.0f / (float)WW);
    }
}

// ---------------- launch ----------------
extern "C" void kernel_launch(void* const* d_in, const int* in_sizes, int n_in,
                              void* d_out, int out_size, void* d_ws, size_t ws_size,
                              hipStream_t stream) {
    const float* F0 = (const float*)d_in[0];
    const float* F1 = (const float*)d_in[1];
    const float* m0 = (const float*)d_in[2];
    const float* m1 = (const float*)d_in[3];
    const float* cw = (const float*)d_in[4];
    const float* cb = (const float*)d_in[5];
    float* out = (float*)d_out;

    // workspace layout (floats)
    const size_t NC  = (size_t)BB * LC * DD;   // per-tensor conv output
    const size_t NBL = (size_t)BB * LC;        // per-tensor row scalars
    float* ws  = (float*)d_ws;
    float* c0  = ws;
    float* c1  = ws + NC;
    float* sq0 = ws + 2 * NC;
    float* sq1 = ws + 2 * NC + NBL;
    float* a0  = ws + 2 * NC + 2 * NBL;
    float* a1  = ws + 2 * NC + 3 * NBL;        // a0,a1 contiguous

    (void)in_sizes; (void)n_in; (void)out_size; (void)ws_size;

    // 1) zero gate accumulators (a0 and a1 are contiguous)
    {
        int n = (int)(2 * NBL);
        zero_kernel<<<(n + 255) / 256, 256, 0, stream>>>(a0, n);
    }
    // 2) conv + tanh + squared norms
    {
        dim3 grid(LC, BB, 2);
        conv_tanh_kernel<<<grid, 256, 0, stream>>>(F0, F1, m0, m1, cw, cb,
                                                   c0, c1, sq0, sq1);
    }
    // 3) WMMA match scores + gate sums (async double-buffered staging)
    {
        dim3 grid((LC + TILE - 1) / TILE, (LC + TILE - 1) / TILE, BB);
        match_kernel<<<grid, 256, 0, stream>>>(c0, c1, sq0, sq1, a0, a1);
    }
    // 4) gated average pool -> outputs (out0 then out1, concatenated)
    {
        dim3 grid(LL, BB, 2);
        pool_out_kernel<<<grid, 256, 0, stream>>>(c0, c1, a0, a1, out);
    }
}